// MimiStyleRVQ_48567490183831
// MI455X (gfx1250) — compile-verified
//
#include <hip/hip_runtime.h>

typedef __attribute__((ext_vector_type(2))) float v2f;
typedef __attribute__((ext_vector_type(4))) float v4f;
typedef __attribute__((ext_vector_type(8))) float v8f;

#define B_  8
#define C_  512
#define T_  12000
#define D_  32
#define K_  1024
#define NC_ 32
#define BT_ (B_ * T_)   // 96000

#define CHUNK_ 256      // codes staged in LDS per pass (4 passes over K=1024)
#define LSTR_  36       // LDS row stride in floats: 8B-aligned pairs, and
                        // lower-half banks = 0 mod 4, upper-half = 2 mod 4
                        // -> conflict-free ds_load_b64 for B operands

// ---------------------------------------------------------------------------
// Projection: x[bt][d] = sum_c latent[b][c][t] * W[d][c] + bias[d]
// One wave = 16-row (t) tile, D=32 -> two 16-wide N tiles, K=512 in steps of
// 4 via V_WMMA_F32_16X16X4_F32. Writes X (kept) and R (residual).
// ---------------------------------------------------------------------------
__global__ __launch_bounds__(128) void proj_kernel(
    const float* __restrict__ latent, const float* __restrict__ W,
    const float* __restrict__ bias, float* __restrict__ X, float* __restrict__ R)
{
  const int lane = threadIdx.x & 31;
  const int wave = threadIdx.x >> 5;
  const int tile = blockIdx.x * 4 + wave;           // 0..5999 (T_/16 per batch)
  const int m    = lane & 15;
  const int half = lane >> 4;
  const int b    = tile / (T_ / 16);
  const int t0   = (tile % (T_ / 16)) * 16;
  const int t    = t0 + m;
  const float* Lb = latent + (size_t)b * C_ * T_;

  v8f c0 = {};  // d 0..15
  v8f c1 = {};  // d 16..31
  #pragma unroll 8
  for (int kk = 0; kk < C_ / 4; ++kk) {
    const int cidx = kk * 4 + 2 * half;
    v2f a;
    a.x = Lb[(size_t)cidx * T_ + t];
    a.y = Lb[(size_t)(cidx + 1) * T_ + t];
    v2f b0 = *(const v2f*)(W + m * C_ + cidx);
    v2f b1 = *(const v2f*)(W + (16 + m) * C_ + cidx);
    c0 = __builtin_amdgcn_wmma_f32_16x16x4_f32(false, a, false, b0, (short)0, c0, false, false);
    c1 = __builtin_amdgcn_wmma_f32_16x16x4_f32(false, a, false, b1, (short)0, c1, false, false);
  }

  const float bias0 = bias[m];
  const float bias1 = bias[16 + m];
  const int bt0 = tile * 16;
  #pragma unroll
  for (int j = 0; j < 8; ++j) {
    const int row = j + 8 * half;
    const float x0 = c0[j] + bias0;
    const float x1 = c1[j] + bias1;
    const size_t o = (size_t)(bt0 + row) * D_;
    X[o + m]      = x0;  R[o + m]      = x0;
    X[o + 16 + m] = x1;  R[o + 16 + m] = x1;
  }
}

// ---------------------------------------------------------------------------
// Codebook squared norms: cbn2[nc*K + k] = sum_d cb[nc][k][d]^2
// ---------------------------------------------------------------------------
__global__ __launch_bounds__(256) void cbnorm_kernel(
    const float* __restrict__ CB, float* __restrict__ cbn2)
{
  const int i = blockIdx.x * 256 + threadIdx.x;   // 0..NC_*K_-1
  const v4f* p = (const v4f*)(CB + (size_t)i * D_);
  float s = 0.f;
  #pragma unroll
  for (int j = 0; j < D_ / 4; ++j) {
    v4f v = p[j];
    s += v.x * v.x + v.y * v.y + v.z * v.z + v.w * v.w;
  }
  cbn2[i] = s;
}

// ---------------------------------------------------------------------------
// One RVQ step (codebook nc). Block = 8 waves; each wave owns 32 residual
// rows (two 16x32 A tiles kept in registers). Codebook is staged to LDS in
// 4 chunks of 256 codes (36 KB, padded rows -> conflict-free ds_load_b64),
// shared by all 8 waves. score[k] = cbn2[k] - 2*dot(r,cb[k]) via f32 WMMA,
// per-row argmin (lowest-index tiebreak), then residual update.
// ---------------------------------------------------------------------------
__global__ __launch_bounds__(256) void rvq_step_kernel(
    float* __restrict__ R, const float* __restrict__ CB,
    const float* __restrict__ cbn2, int* __restrict__ idxOut, int nc)
{
  __shared__ float cbS[CHUNK_ * LSTR_];           // 36864 bytes

  const int tid  = threadIdx.x;
  const int lane = tid & 31;
  const int wave = tid >> 5;
  const int tile = blockIdx.x * 8 + wave;         // 0..2999 (32-row tiles)
  const int bt0  = tile * 32;
  const int m    = lane & 15;
  const int half = lane >> 4;
  const int kb   = 2 * half;                      // K offset base for this lane half

  const float* CBn = CB + (size_t)nc * K_ * D_;
  const float* cn2 = cbn2 + (size_t)nc * K_;

  // A operands: rows bt0+m (tile0) and bt0+16+m (tile1), loaded once.
  const float* r0 = R + (size_t)(bt0 + m) * D_;
  const float* r1 = R + (size_t)(bt0 + 16 + m) * D_;
  v2f a0[8], a1[8];
  #pragma unroll
  for (int kk = 0; kk < 8; ++kk) {
    a0[kk] = *(const v2f*)(r0 + kk * 4 + kb);
    a1[kk] = *(const v2f*)(r1 + kk * 4 + kb);
  }

  float bS0[8], bS1[8];
  int   bI0[8], bI1[8];
  #pragma unroll
  for (int j = 0; j < 8; ++j) {
    bS0[j] = __builtin_inff(); bI0[j] = 0;
    bS1[j] = __builtin_inff(); bI1[j] = 0;
  }

  for (int chunk = 0; chunk < K_ / CHUNK_; ++chunk) {
    __syncthreads();                              // all waves done with prev chunk
    // Cooperative stage: 256 codes x 32 floats as 64-bit pairs.
    // Global side coalesced (consecutive pairs); LDS stores near conflict-free.
    const float* src = CBn + (size_t)chunk * CHUNK_ * D_;
    #pragma unroll
    for (int i = 0; i < 16; ++i) {
      const int q    = tid + i * 256;             // 0..4095 pair index
      const int code = q >> 4;
      const int p    = q & 15;
      v2f v = *(const v2f*)(src + code * D_ + p * 2);
      *(v2f*)(cbS + code * LSTR_ + p * 2) = v;
    }
    __syncthreads();

    for (int nt = 0; nt < CHUNK_ / 16; ++nt) {
      const int codeL = nt * 16 + m;
      const int code  = chunk * CHUNK_ + codeL;
      const float* crow = cbS + codeL * LSTR_ + kb;
      v8f c0 = {};
      v8f c1 = {};
      #pragma unroll
      for (int kk = 0; kk < 8; ++kk) {
        v2f bv = *(const v2f*)(crow + kk * 4);    // ds_load_b64, conflict-free
        c0 = __builtin_amdgcn_wmma_f32_16x16x4_f32(false, a0[kk], false, bv, (short)0, c0, false, false);
        c1 = __builtin_amdgcn_wmma_f32_16x16x4_f32(false, a1[kk], false, bv, (short)0, c1, false, false);
      }
      const float c2v = cn2[code];
      #pragma unroll
      for (int j = 0; j < 8; ++j) {
        const float s0 = c2v - 2.0f * c0[j];
        if (s0 < bS0[j]) { bS0[j] = s0; bI0[j] = code; }
        const float s1 = c2v - 2.0f * c1[j];
        if (s1 < bS1[j]) { bS1[j] = s1; bI1[j] = code; }
      }
    }
  }

  // Argmin across the 16 lanes of each half (rows m=j in lanes 0-15,
  // m=j+8 in lanes 16-31). Tiebreak: lowest index (matches jnp.argmin).
  #pragma unroll
  for (int j = 0; j < 8; ++j) {
    float s0 = bS0[j]; int i0 = bI0[j];
    float s1 = bS1[j]; int i1 = bI1[j];
    #pragma unroll
    for (int off = 1; off < 16; off <<= 1) {
      const float t0 = __shfl_xor(s0, off, 32);
      const int   u0 = __shfl_xor(i0, off, 32);
      if (t0 < s0 || (t0 == s0 && u0 < i0)) { s0 = t0; i0 = u0; }
      const float t1 = __shfl_xor(s1, off, 32);
      const int   u1 = __shfl_xor(i1, off, 32);
      if (t1 < s1 || (t1 == s1 && u1 < i1)) { s1 = t1; i1 = u1; }
    }
    bI0[j] = i0;                                  // uniform within each 16-lane group
    bI1[j] = i1;
  }

  if (m == 0) {
    #pragma unroll
    for (int j = 0; j < 8; ++j) {
      idxOut[nc * BT_ + bt0 + j + 8 * half]      = bI0[j];
      idxOut[nc * BT_ + bt0 + 16 + j + 8 * half] = bI1[j];
    }
  }

  // residual -= chosen code (32 lanes update one 32-float row per pass)
  #pragma unroll
  for (int j = 0; j < 8; ++j) {
    const int i0a = __shfl(bI0[j], 0, 32);        // row j
    const int i0b = __shfl(bI0[j], 16, 32);       // row j+8
    const int i1a = __shfl(bI1[j], 0, 32);        // row 16+j
    const int i1b = __shfl(bI1[j], 16, 32);       // row 24+j
    R[(size_t)(bt0 + j) * D_ + lane]      -= CBn[(size_t)i0a * D_ + lane];
    R[(size_t)(bt0 + 8 + j) * D_ + lane]  -= CBn[(size_t)i0b * D_ + lane];
    R[(size_t)(bt0 + 16 + j) * D_ + lane] -= CBn[(size_t)i1a * D_ + lane];
    R[(size_t)(bt0 + 24 + j) * D_ + lane] -= CBn[(size_t)i1b * D_ + lane];
  }
}

// ---------------------------------------------------------------------------
// out[b][d][t] = X[bt][d] - R_final[bt][d]  (== quant, STE), via LDS transpose
// ---------------------------------------------------------------------------
__global__ __launch_bounds__(256) void finalize_kernel(
    const float* __restrict__ X, const float* __restrict__ R,
    float* __restrict__ out)
{
  __shared__ float tileS[32][33];                 // stride 33 -> bank-conflict free
  const int blk = blockIdx.x;                     // 0..2999
  const int b   = blk / (T_ / 32);
  const int t0  = (blk % (T_ / 32)) * 32;
  const size_t base = ((size_t)b * T_ + t0) * D_;
  const int tid = threadIdx.x;

  #pragma unroll
  for (int it = 0; it < 4; ++it) {
    const int idx = it * 256 + tid;
    const int tl = idx >> 5, d = idx & 31;        // d fastest -> coalesced loads
    const size_t o = base + (size_t)tl * D_ + d;
    tileS[tl][d] = X[o] - R[o];
  }
  __syncthreads();
  #pragma unroll
  for (int it = 0; it < 4; ++it) {
    const int idx = it * 256 + tid;
    const int d = idx >> 5, tl = idx & 31;        // t fastest -> coalesced stores
    out[(size_t)b * D_ * T_ + (size_t)d * T_ + t0 + tl] = tileS[tl][d];
  }
}

// ---------------------------------------------------------------------------
extern "C" void kernel_launch(void* const* d_in, const int* in_sizes, int n_in,
                              void* d_out, int out_size, void* d_ws, size_t ws_size,
                              hipStream_t stream)
{
  (void)in_sizes; (void)n_in; (void)out_size; (void)ws_size;
  const float* latent = (const float*)d_in[0];   // [B, C, T]
  const float* W      = (const float*)d_in[1];   // [D, C]
  const float* bias   = (const float*)d_in[2];   // [D]
  const float* CB     = (const float*)d_in[3];   // [NC, K, D]

  float* X    = (float*)d_ws;                    // [BT, D]  projected x (kept)
  float* R    = X + (size_t)BT_ * D_;            // [BT, D]  residual
  float* cbn2 = R + (size_t)BT_ * D_;            // [NC, K]  codebook norms

  float* outF = (float*)d_out;                   // [B, D, T] f32
  int*   outI = (int*)d_out + (size_t)B_ * D_ * T_;  // [NC, B, T] int32

  proj_kernel<<<BT_ / (16 * 4), 128, 0, stream>>>(latent, W, bias, X, R);
  cbnorm_kernel<<<(NC_ * K_) / 256, 256, 0, stream>>>(CB, cbn2);
  for (int nc = 0; nc < NC_; ++nc)
    rvq_step_kernel<<<BT_ / (32 * 8), 256, 0, stream>>>(R, CB, cbn2, outI, nc);
  finalize_kernel<<<(B_ * T_) / 32, 256, 0, stream>>>(X, R, outF);
}